// DotDecoder_7052336300554
// MI455X (gfx1250) — compile-verified
//
#include <hip/hip_runtime.h>

typedef __attribute__((ext_vector_type(2))) float v2f;
typedef __attribute__((ext_vector_type(4))) float v4f;
typedef __attribute__((ext_vector_type(8))) float v8f;

// 128-bit payload type for the async global->LDS builtin (gcc-style vector,
// matching the builtin's declared parameter type from the compiler diagnostic).
typedef int v4i_vs __attribute__((vector_size(16)));
typedef __attribute__((address_space(1))) v4i_vs* gbl_b128_ptr;
typedef __attribute__((address_space(3))) v4i_vs* lds_b128_ptr;

constexpr int kN       = 100;       // graph size
constexpr int kD       = 128;       // feature dim
constexpr int kNPad    = 112;       // 7 * 16 (tile-padded rows)
constexpr int kNT      = 7;         // tiles per dimension
constexpr int kXStride = kD + 4;    // 132 floats: LDS bank-conflict-free A/B loads
constexpr int kSStride = 104;       // padded score row stride
constexpr int kThreads = 256;       // 8 wave32 per workgroup

#define HAS_ASYNC_LDS __has_builtin(__builtin_amdgcn_global_load_async_to_lds_b128)

__global__ __launch_bounds__(kThreads)
void dot_softmax_kernel(const float* __restrict__ x, float* __restrict__ out) {
  __shared__ float x_s[kNPad * kXStride];   // 59,136 B
  __shared__ float s_s[kN * kSStride];      // 41,600 B
  __shared__ float pmax_s[2 * kN];          // partial column max
  __shared__ float psum_s[2 * kN];          // partial column sum(exp)
  __shared__ float colmax[128];
  __shared__ float colrcp[128];

  const int g   = blockIdx.x;
  const int tid = threadIdx.x;
  const float* __restrict__ xg = x + (size_t)g * kN * kD;

  // ---- Stage X[g] (100x128 f32) into LDS with padded row stride ----
#if HAS_ASYNC_LDS
  // CDNA5 async path: memory -> LDS directly, tracked by ASYNCcnt.
  for (int idx = tid; idx < kN * (kD / 4); idx += kThreads) {
    const int row = idx >> 5;          // kD/4 == 32 float4 per row
    const int c4  = idx & 31;
    __builtin_amdgcn_global_load_async_to_lds_b128(
        (gbl_b128_ptr)(xg + row * kD + c4 * 4),
        (lds_b128_ptr)(&x_s[row * kXStride + c4 * 4]),
        /*offset=*/0, /*cpol=*/0);
  }
#else
  for (int idx = tid; idx < kN * (kD / 4); idx += kThreads) {
    const int row = idx >> 5;
    const int c4  = idx & 31;
    v4f v = *(const v4f*)(xg + row * kD + c4 * 4);
    *(v4f*)(&x_s[row * kXStride + c4 * 4]) = v;
  }
#endif
  // Zero tile-pad rows 100..111 so WMMA needs no masking.
  for (int idx = tid; idx < (kNPad - kN) * kXStride; idx += kThreads) {
    x_s[kN * kXStride + idx] = 0.0f;
  }
#if HAS_ASYNC_LDS
#if __has_builtin(__builtin_amdgcn_s_wait_asynccnt)
  __builtin_amdgcn_s_wait_asynccnt(0);
#else
  asm volatile("s_wait_asynccnt 0" ::: "memory");
#endif
#endif
  __syncthreads();

  // ---- S = X * X^T via V_WMMA_F32_16X16X4_F32 over 7x7 tiles of 16x16 ----
  const int wid  = tid >> 5;
  const int lane = tid & 31;
  const int lm   = lane & 15;
  const int lhi  = lane >> 4;          // 0/1 selects K pair {0,1} vs {2,3}

  for (int t = wid; t < kNT * kNT; t += (kThreads / 32)) {   // wave-uniform
    const int m0 = (t / kNT) * 16;
    const int n0 = (t % kNT) * 16;
    // A fragment: row (m0+lm), cols k+2*lhi .. k+2*lhi+1  (16x4 f32 layout)
    // B fragment (= X^T tile): identical addressing with row base n0.
    const float* ap = &x_s[(m0 + lm) * kXStride + 2 * lhi];
    const float* bp = &x_s[(n0 + lm) * kXStride + 2 * lhi];

    v8f c = {};
#pragma unroll
    for (int k = 0; k < kD; k += 4) {
      v2f a = *(const v2f*)(ap + k);   // ds_load_2addr_b64 (paired by compiler)
      v2f b = *(const v2f*)(bp + k);
      c = __builtin_amdgcn_wmma_f32_16x16x4_f32(
          /*neg_a=*/false, a, /*neg_b=*/false, b,
          /*c_mod=*/(short)0, c, /*reuse_a=*/false, /*reuse_b=*/false);
    }

    // C/D layout: lane holds rows r + 8*lhi (VGPR r), column n0+lm.
    const int col = n0 + lm;
    if (col < kN) {
#pragma unroll
      for (int r = 0; r < 8; ++r) {
        const int row = m0 + r + 8 * lhi;
        if (row < kN) s_s[row * kSStride + col] = c[r];
      }
    }
  }
  __syncthreads();

  // ---- Softmax over axis=1 (rows i for each column j), 2-way split rows ----
  if (tid < 2 * kN) {
    const int s  = tid / kN;           // row slice 0/1
    const int j  = tid - s * kN;       // column
    const int i0 = s * (kN / 2);
    const int i1 = i0 + (kN / 2);
    float m = -3.402823466e38f;
    for (int i = i0; i < i1; ++i) m = fmaxf(m, s_s[i * kSStride + j]);
    float sum = 0.0f;
    for (int i = i0; i < i1; ++i) sum += __expf(s_s[i * kSStride + j] - m);
    pmax_s[tid] = m;
    psum_s[tid] = sum;
  }
  __syncthreads();

  if (tid < kN) {
    const int j = tid;
    const float m0 = pmax_s[j], m1 = pmax_s[kN + j];
    const float M  = fmaxf(m0, m1);
    const float sum = psum_s[j] * __expf(m0 - M) + psum_s[kN + j] * __expf(m1 - M);
    colmax[j] = M;
    colrcp[j] = 1.0f / sum;
  }
  __syncthreads();

  // ---- Coalesced normalized writeback: out[g, i, j] ----
  float* __restrict__ og = out + (size_t)g * kN * kN;
  for (int idx = tid; idx < kN * kN; idx += kThreads) {
    const int i = idx / kN;
    const int j = idx - i * kN;
    og[idx] = __expf(s_s[i * kSStride + j] - colmax[j]) * colrcp[j];
  }
}

extern "C" void kernel_launch(void* const* d_in, const int* in_sizes, int n_in,
                              void* d_out, int out_size, void* d_ws, size_t ws_size,
                              hipStream_t stream) {
  (void)n_in; (void)out_size; (void)d_ws; (void)ws_size;
  const float* x = (const float*)d_in[0];     // [B*N, D] f32
  // d_in[1] = edge_index (unused by the math), d_in[2] = graph_size (==100)
  float* out = (float*)d_out;                 // [B, N, N] f32
  const int b = in_sizes[0] / (kN * kD);      // 1024
  dot_softmax_kernel<<<dim3(b), dim3(kThreads), 0, stream>>>(x, out);
}